// GRU_25091198943527
// MI455X (gfx1250) — compile-verified
//
#include <hip/hip_runtime.h>
#include <hip/hip_bf16.h>
#include <math.h>

#define N_NODES 100000
#define N_EDGES 1600000
#define D 128
#define TWO_D 256
#define BN_EPS 1e-5f

typedef float v2f __attribute__((ext_vector_type(2)));
typedef float v8f __attribute__((ext_vector_type(8)));

typedef int vi4 __attribute__((vector_size(16)));
typedef __attribute__((address_space(1))) vi4 gvi4;  // global (AS1)
typedef __attribute__((address_space(3))) vi4 lvi4;  // LDS (AS3)

#if defined(__HIP_DEVICE_COMPILE__) && __has_builtin(__builtin_amdgcn_global_load_async_to_lds_b128)
#define USE_ASYNC_LDS 1
#else
#define USE_ASYNC_LDS 0
#endif

// Copy 16 bytes global -> LDS. Async path uses CDNA5
// GLOBAL_LOAD_ASYNC_TO_LDS_B128 (tracked by ASYNCcnt); fallback is a
// regular float4 round-trip through VGPRs.
__device__ __forceinline__ void stage16(void* lds_dst, const void* gsrc) {
#if USE_ASYNC_LDS
  __builtin_amdgcn_global_load_async_to_lds_b128(
      (gvi4*)gsrc, (lvi4*)lds_dst, 0, 0);
#else
  *(float4*)lds_dst = *(const float4*)gsrc;
#endif
}

__device__ __forceinline__ void stage_wait() {
#if USE_ASYNC_LDS
#if __has_builtin(__builtin_amdgcn_s_wait_asynccnt)
  __builtin_amdgcn_s_wait_asynccnt(0);
#else
  asm volatile("s_wait_asynccnt 0x0" ::: "memory");
#endif
#endif
}

__device__ __forceinline__ void atomAddF32(float* p, float v) {
  // native global_atomic_add_f32 (no CAS loop)
  unsafeAtomicAdd(p, v);
}

// ---------------- BatchNorm: column sums ----------------
__global__ __launch_bounds__(128) void bn_stats_kernel(
    const float* __restrict__ feat, float* __restrict__ ssum, float* __restrict__ ssq) {
  const int col = threadIdx.x;  // 0..127
  float s = 0.f, s2 = 0.f;
  for (int r = blockIdx.x; r < N_NODES; r += gridDim.x) {
    float v = feat[(size_t)r * D + col];
    s += v;
    s2 += v * v;
  }
  atomAddF32(&ssum[col], s);
  atomAddF32(&ssq[col], s2);
}

// ---------------- BatchNorm: finalize scale/shift ----------------
__global__ __launch_bounds__(128) void bn_finalize_kernel(
    const float* __restrict__ ssum, const float* __restrict__ ssq,
    const float* __restrict__ gamma, const float* __restrict__ beta,
    float* __restrict__ scale, float* __restrict__ shift) {
  const int c = threadIdx.x;
  const float invN = 1.0f / (float)N_NODES;
  float mu = ssum[c] * invN;
  float var = ssq[c] * invN - mu * mu;  // biased variance (matches jnp.var)
  float sc = rsqrtf(var + BN_EPS) * gamma[c];
  scale[c] = sc;
  shift[c] = beta[c] - mu * sc;
}

// ---------------- BatchNorm: normalize ----------------
__global__ __launch_bounds__(256) void bn_norm_kernel(
    const float* __restrict__ feat, const float* __restrict__ scale,
    const float* __restrict__ shift, float* __restrict__ x) {
  __shared__ float ssc[D];
  __shared__ float ssh[D];
  if (threadIdx.x < D) {
    ssc[threadIdx.x] = scale[threadIdx.x];
    ssh[threadIdx.x] = shift[threadIdx.x];
  }
  __syncthreads();
  const int total4 = N_NODES * (D / 4);
  for (int i = blockIdx.x * blockDim.x + threadIdx.x; i < total4;
       i += gridDim.x * blockDim.x) {
    const int c4 = (i & 31) * 4;  // D/4 == 32
    float4 v = ((const float4*)feat)[i];
    float4 o;
    o.x = v.x * ssc[c4 + 0] + ssh[c4 + 0];
    o.y = v.y * ssc[c4 + 1] + ssh[c4 + 1];
    o.z = v.z * ssc[c4 + 2] + ssh[c4 + 2];
    o.w = v.w * ssc[c4 + 3] + ssh[c4 + 3];
    ((float4*)x)[i] = o;
  }
}

// ---------------- GEMM: out[N,128] = x[N,128] @ W[128,128] + bias ----------------
// block: 256 threads = 8 waves, computes 32 rows x 64 cols of output.
__global__ __launch_bounds__(256) void gemm_xw_kernel(
    const float* __restrict__ X, const float* __restrict__ W,
    const float* __restrict__ bias, float* __restrict__ out) {
  __shared__ float sx[32][132];   // 32 x 128 x-tile (padded, rows 16B-aligned)
  __shared__ float sw[128][68];   // 128 x 64 W-panel (padded, rows 16B-aligned)
  const int tid = threadIdx.x;
  const int rowBase = blockIdx.x * 32;
  const int colBase = blockIdx.y * 64;

  for (int i = tid; i < 32 * 32; i += 256) {
    const int r = i >> 5, c4 = (i & 31) * 4;
    stage16(&sx[r][c4], &X[(size_t)(rowBase + r) * D + c4]);
  }
  for (int i = tid; i < 128 * 16; i += 256) {
    const int r = i >> 4, c4 = (i & 15) * 4;
    stage16(&sw[r][c4], &W[(size_t)r * D + colBase + c4]);
  }
  stage_wait();
  __syncthreads();

  const int wv = tid >> 5, lane = tid & 31;
  const int wRow = (wv >> 2) * 16;         // 0 or 16
  const int wCol = (wv & 3) * 16;          // 0,16,32,48
  const int ml = lane & 15;
  const int m = wRow + ml;                 // A row for this lane
  const int nc = wCol + ml;                // B col for this lane (within panel)
  const int kh = (lane >= 16) ? 2 : 0;     // K half select per VGPR layout
  const int rAdd = (lane >= 16) ? 8 : 0;   // C/D layout: lanes 16-31 hold M+8

  v8f acc = {};
  for (int k = 0; k < D; k += 4) {
    v2f A, B;
    A.x = sx[m][k + kh];     A.y = sx[m][k + kh + 1];
    B.x = sw[k + kh][nc];    B.y = sw[k + kh + 1][nc];
    acc = __builtin_amdgcn_wmma_f32_16x16x4_f32(false, A, false, B,
                                                (short)0, acc, false, false);
  }
  const int col = colBase + nc;
  const float bcol = bias[col];
#pragma unroll
  for (int j = 0; j < 8; ++j) {
    const int row = rowBase + wRow + j + rAdd;
    out[(size_t)row * D + col] = acc[j] + bcol;
  }
}

// ---------------- Edge scatter: aout[sidx[e]] += featp[gidx[e]] * ew[e] ----------------
// one wave per edge; lane handles 4 contiguous floats.
__global__ __launch_bounds__(256) void scatter_kernel(
    const float* __restrict__ featp, const int* __restrict__ gidx,
    const int* __restrict__ sidx, const float* __restrict__ ew,
    float* __restrict__ aout) {
  const int e = blockIdx.x * 8 + (threadIdx.x >> 5);
  if (e >= N_EDGES) return;
  const int lane = threadIdx.x & 31;
  const int g = gidx[e];
  const int s = sidx[e];
  const float wt = ew[e];
  float4 v = *(const float4*)&featp[(size_t)g * D + lane * 4];
  float* dp = &aout[(size_t)s * TWO_D + lane * 4];
  atomAddF32(dp + 0, v.x * wt);
  atomAddF32(dp + 1, v.y * wt);
  atomAddF32(dp + 2, v.z * wt);
  atomAddF32(dp + 3, v.w * wt);
}

// ---------------- Gate: f = a@Wg_in + bg; b = x@Wg_out; fused gating ----------------
// block: 256 threads = 8 waves, handles 16 rows.
__global__ __launch_bounds__(256) void gate_kernel(
    const float* __restrict__ A_, const float* __restrict__ X,
    const float* __restrict__ Wg_in, const float* __restrict__ bg_in,
    const float* __restrict__ Wg_out, float* __restrict__ out) {
  __shared__ float sa[16][260];  // 16 x 256 a-tile (rows 16B-aligned)
  __shared__ float sxg[16][132]; // 16 x 128 x-tile
  __shared__ float sf[16][260];  // f result
  __shared__ float sb[16][260];  // b result
  const int tid = threadIdx.x;
  const int rowBase = blockIdx.x * 16;

  for (int i = tid; i < 16 * 64; i += 256) {
    const int r = i >> 6, c4 = (i & 63) * 4;
    stage16(&sa[r][c4], &A_[(size_t)(rowBase + r) * TWO_D + c4]);
  }
  for (int i = tid; i < 16 * 32; i += 256) {
    const int r = i >> 5, c4 = (i & 31) * 4;
    stage16(&sxg[r][c4], &X[(size_t)(rowBase + r) * D + c4]);
  }
  stage_wait();
  __syncthreads();

  const int wv = tid >> 5, lane = tid & 31;
  const int ml = lane & 15;
  const int kh = (lane >= 16) ? 2 : 0;
  const int rAdd = (lane >= 16) ? 8 : 0;

  // f = a @ Wg_in + bg_in  (K = 256), each wave does 2 of 16 col-tiles
  for (int t = wv; t < 16; t += 8) {
    const int col = t * 16 + ml;
    v8f acc = {};
    for (int k = 0; k < TWO_D; k += 4) {
      v2f A, B;
      A.x = sa[ml][k + kh];  A.y = sa[ml][k + kh + 1];
      B.x = Wg_in[(size_t)(k + kh) * TWO_D + col];
      B.y = Wg_in[(size_t)(k + kh + 1) * TWO_D + col];
      acc = __builtin_amdgcn_wmma_f32_16x16x4_f32(false, A, false, B,
                                                  (short)0, acc, false, false);
    }
    const float bcol = bg_in[col];
#pragma unroll
    for (int j = 0; j < 8; ++j) sf[j + rAdd][col] = acc[j] + bcol;
  }

  // b = x @ Wg_out  (K = 128)
  for (int t = wv; t < 16; t += 8) {
    const int col = t * 16 + ml;
    v8f acc = {};
    for (int k = 0; k < D; k += 4) {
      v2f A, B;
      A.x = sxg[ml][k + kh]; A.y = sxg[ml][k + kh + 1];
      B.x = Wg_out[(size_t)(k + kh) * TWO_D + col];
      B.y = Wg_out[(size_t)(k + kh + 1) * TWO_D + col];
      acc = __builtin_amdgcn_wmma_f32_16x16x4_f32(false, A, false, B,
                                                  (short)0, acc, false, false);
    }
#pragma unroll
    for (int j = 0; j < 8; ++j) sb[j + rAdd][col] = acc[j];
  }
  __syncthreads();

  // gating: out = new_gate + input_gate * (x - new_gate)
  for (int i = tid; i < 16 * D; i += 256) {
    const int r = i >> 7, c = i & 127;
    const float ig = 1.0f / (1.0f + expf(-(sf[r][c] + sb[r][c])));
    const float ng = tanhf(sf[r][c + 128] + sb[r][c + 128]);
    const float xv = sxg[r][c];
    out[(size_t)(rowBase + r) * D + c] = ng + ig * (xv - ng);
  }
}

extern "C" void kernel_launch(void* const* d_in, const int* in_sizes, int n_in,
                              void* d_out, int out_size, void* d_ws, size_t ws_size,
                              hipStream_t stream) {
  const float* feat  = (const float*)d_in[0];
  const int*   src   = (const int*)d_in[1];
  const int*   dst   = (const int*)d_in[2];
  const float* iw    = (const float*)d_in[3];
  const float* ow    = (const float*)d_in[4];
  const float* gamma = (const float*)d_in[5];
  const float* beta  = (const float*)d_in[6];
  const float* W_in  = (const float*)d_in[7];
  const float* b_in  = (const float*)d_in[8];
  const float* W_out = (const float*)d_in[9];
  const float* b_out = (const float*)d_in[10];
  const float* Wg_in = (const float*)d_in[11];
  const float* bg_in = (const float*)d_in[12];
  const float* Wg_out= (const float*)d_in[13];
  float* out = (float*)d_out;

  float* ws = (float*)d_ws;
  const size_t ND = (size_t)N_NODES * D;
  float* x        = ws;             // [N,128]
  float* feat_in  = ws + ND;        // [N,128]
  float* feat_out = ws + 2 * ND;    // [N,128]
  float* a        = ws + 3 * ND;    // [N,256]  (a_in | a_out)
  float* stats    = ws + 5 * ND;    // sums(128) sq(128) scale(128) shift(128)
  float* ssum  = stats;
  float* ssq   = stats + D;
  float* scale = stats + 2 * D;
  float* shift = stats + 3 * D;

  (void)hipMemsetAsync(stats, 0, 2 * D * sizeof(float), stream);
  (void)hipMemsetAsync(a, 0, (size_t)N_NODES * TWO_D * sizeof(float), stream);

  bn_stats_kernel<<<512, 128, 0, stream>>>(feat, ssum, ssq);
  bn_finalize_kernel<<<1, 128, 0, stream>>>(ssum, ssq, gamma, beta, scale, shift);
  bn_norm_kernel<<<2048, 256, 0, stream>>>(feat, scale, shift, x);

  gemm_xw_kernel<<<dim3(N_NODES / 32, 2), 256, 0, stream>>>(x, W_in, b_in, feat_in);
  gemm_xw_kernel<<<dim3(N_NODES / 32, 2), 256, 0, stream>>>(x, W_out, b_out, feat_out);

  // a_in  = segsum(feat_in[src] * iw, dst) -> cols [0,128)
  scatter_kernel<<<N_EDGES / 8, 256, 0, stream>>>(feat_in, src, dst, iw, a);
  // a_out = segsum(feat_out[dst] * ow, src) -> cols [128,256)
  scatter_kernel<<<N_EDGES / 8, 256, 0, stream>>>(feat_out, dst, src, ow, a + D);

  gate_kernel<<<N_NODES / 16, 256, 0, stream>>>(a, x, Wg_in, bg_in, Wg_out, out);
}